// MS_WSA_9543417332196
// MI455X (gfx1250) — compile-verified
//
#include <hip/hip_runtime.h>
#include <math.h>
#include <stdint.h>

// ---------------- problem constants (from reference setup) ----------------
#define NWIN  2048
#define PWIN  64
#define KTOK  48
#define KNP   40      // K - PAD, non-padding tokens per window
#define MSEL  1536
#define CCH   128
#define HEADS 4
#define DHEAD 32
#define GINNER 320
#define BSZ   8

typedef __attribute__((ext_vector_type(2))) float v2f;
typedef __attribute__((ext_vector_type(8))) float v8f;
typedef __attribute__((ext_vector_type(4))) unsigned int v4u;
typedef __attribute__((ext_vector_type(8))) int v8i_;
typedef __attribute__((ext_vector_type(4))) int v4i_;

#if defined(__has_builtin)
#if __has_builtin(__builtin_amdgcn_tensor_load_to_lds) && \
    __has_builtin(__builtin_amdgcn_s_wait_tensorcnt)
#define USE_TDM 1
#endif
#endif
#ifndef USE_TDM
#define USE_TDM 0
#endif

// ---------------- WMMA f32 16x16x4 helpers ----------------
__device__ __forceinline__ v8f wmma4(v2f a, v2f b, v8f c) {
  // D = A(16x4) * B(4x16) + C, fp32 end to end
  return __builtin_amdgcn_wmma_f32_16x16x4_f32(false, a, false, b, (short)0, c,
                                               false, false);
}

// A fragment from row-major matrix (pre-offset to tile row0), lda in floats.
// Layout per ISA 32-bit A 16x4: VGPR0 = K=k0 (lanes0-15) | K=k0+2 (lanes16-31),
// VGPR1 = K=k0+1 | k0+3.  Also reused for B^T (B[k][n] = src[n][k]).
__device__ __forceinline__ v2f frag_rm(const float* A, int lda, int k0, int lane) {
  int half = lane >> 4, l = lane & 15;
  v2f a;
  a.x = A[l * lda + k0 + 2 * half];
  a.y = A[l * lda + k0 + 2 * half + 1];
  return a;
}
// B fragment from row-major K x N matrix (pre-offset to column n0).
__device__ __forceinline__ v2f frag_b(const float* B, int ldb, int k0, int lane) {
  int half = lane >> 4, l = lane & 15;
  v2f b;
  b.x = B[(k0 + 2 * half) * ldb + l];
  b.y = B[(k0 + 2 * half + 1) * ldb + l];
  return b;
}
// C/D 16x16: VGPR r holds row r (lanes0-15) / row r+8 (lanes16-31), col = lane&15.
__device__ __forceinline__ void store_c(float* D, int ldd, v8f c, int lane) {
  int half = lane >> 4, l = lane & 15;
#pragma unroll
  for (int r = 0; r < 8; ++r) D[(r + 8 * half) * ldd + l] = c[r];
}

__device__ __forceinline__ float wsum(float v) {
#pragma unroll
  for (int o = 16; o > 0; o >>= 1) v += __shfl_xor(v, o, 32);
  return v;
}
__device__ __forceinline__ float gelu_exact(float x) {
  return 0.5f * x * (1.0f + erff(x * 0.70710678118654752440f));
}

// ---------------- Tensor Data Mover: 2D f32 tile global -> LDS ----------------
// Builds a 2-group D# (count=1, data_size=4B, type=2) per CDNA5 ISA ch.8 and
// issues TENSOR_LOAD_TO_LDS; completion tracked with TENSORcnt.
#if USE_TDM
__device__ __forceinline__ void tdm_load2d_f32(void* lds_dst, const float* gsrc,
                                               unsigned rows, unsigned cols,
                                               unsigned stride_elems) {
  unsigned lds_off = (unsigned)(unsigned long long)(uintptr_t)lds_dst;
  unsigned long long ga = (unsigned long long)(uintptr_t)gsrc;
  v4u g0;
  g0[0] = 1u;                                            // count=1 valid user D#
  g0[1] = lds_off;                                       // lds_addr (bytes)
  g0[2] = (unsigned)ga;                                  // global_addr[31:0]
  g0[3] = (unsigned)((ga >> 32) & 0x01FFFFFFu) | (2u << 30);  // addr[56:32]|type=2
  v8i_ g1;
  g1[0] = (int)(2u << 16);                               // data_size=4B, mask=0
  g1[1] = (int)((cols & 0xFFFFu) << 16);                 // tensor_dim0[15:0]@63:48
  g1[2] = (int)(((cols >> 16) & 0xFFFFu) |               // tensor_dim0[31:16]
                ((rows & 0xFFFFu) << 16));               // tensor_dim1[15:0]
  g1[3] = (int)(((rows >> 16) & 0xFFFFu) |               // tensor_dim1[31:16]
                ((cols & 0xFFFFu) << 16));               // tile_dim0
  g1[4] = (int)(rows & 0xFFFFu);                         // tile_dim1 (tile_dim2=0)
  g1[5] = (int)stride_elems;                             // tensor_dim0_stride lo32
  g1[6] = 0;                                             // stride hi / dim1_stride
  g1[7] = 0;
  v4i_ gz;
  gz[0] = 0; gz[1] = 0; gz[2] = 0; gz[3] = 0;
#if defined(__clang_major__) && (__clang_major__ >= 23)
  v8i_ gz8;
#pragma unroll
  for (int i = 0; i < 8; ++i) gz8[i] = 0;
  __builtin_amdgcn_tensor_load_to_lds(g0, g1, gz, gz, gz8, 0);
#else
  __builtin_amdgcn_tensor_load_to_lds(g0, g1, gz, gz, 0);
#endif
}
#endif

// Stage a contiguous rows x cols f32 tile into LDS (TDM if available).
// Must be followed by __syncthreads() at the call site.
__device__ __forceinline__ void stage_tile(float* sdst, const float* gsrc,
                                           unsigned rows, unsigned cols) {
#if USE_TDM
  if ((threadIdx.x >> 5) == 0) {          // one wave issues the DMA descriptor
    tdm_load2d_f32(sdst, gsrc, rows, cols, cols);
    __builtin_amdgcn_s_wait_tensorcnt(0); // TENSORcnt -> 0 before barrier
  }
#else
  for (unsigned i = threadIdx.x; i < rows * cols / 4; i += blockDim.x)
    ((float4*)sdst)[i] = ((const float4*)gsrc)[i];
#endif
}

// ---------------- kernel 1a: LN1 over all tokens -> d_out ----------------
__global__ void k_ln1(const float* __restrict__ x, const float* __restrict__ g,
                      const float* __restrict__ b, float* __restrict__ out) {
  int wave = threadIdx.x >> 5, lane = threadIdx.x & 31;
  long long row = (long long)blockIdx.x * 8 + wave;
  const float* xr = x + row * CCH;
  float v[4];
  float s = 0.f, ss = 0.f;
#pragma unroll
  for (int i = 0; i < 4; ++i) {
    v[i] = xr[lane * 4 + i];
    s += v[i];
    ss += v[i] * v[i];
  }
  s = wsum(s);
  ss = wsum(ss);
  float mean = s * (1.0f / CCH);
  float var = ss * (1.0f / CCH) - mean * mean;
  float rstd = rsqrtf(var + 1e-5f);
  float* orow = out + row * CCH;
#pragma unroll
  for (int i = 0; i < 4; ++i) {
    int c = lane * 4 + i;
    orow[c] = (v[i] - mean) * rstd * g[c] + b[c];
  }
}

// ------ kernel 1b: gather selected windows; LN2 on kk<40 -> XT (=xt, shortcut) ------
__global__ void k_build_xt(const float* __restrict__ X, const int* __restrict__ iw,
                           const float* __restrict__ g2, const float* __restrict__ b2,
                           float* __restrict__ XT) {
  int wave = threadIdx.x >> 5, lane = threadIdx.x & 31;
  int row = blockIdx.x * 8 + wave;  // 0 .. MSEL*KTOK-1
  int m = row / KTOK, kk = row - m * KTOK;
  int w = iw[m];
  const float* src = X + ((long long)w * PWIN + kk) * CCH;
  float* dst = XT + (long long)row * CCH;
  if (kk < KNP) {
    float v[4];
    float s = 0.f, ss = 0.f;
#pragma unroll
    for (int i = 0; i < 4; ++i) {
      v[i] = src[lane * 4 + i];
      s += v[i];
      ss += v[i] * v[i];
    }
    s = wsum(s);
    ss = wsum(ss);
    float mean = s * (1.0f / CCH);
    float var = ss * (1.0f / CCH) - mean * mean;
    float rstd = rsqrtf(var + 1e-5f);
#pragma unroll
    for (int i = 0; i < 4; ++i) {
      int c = lane * 4 + i;
      dst[c] = (v[i] - mean) * rstd * g2[c] + b2[c];
    }
  } else {
#pragma unroll
    for (int i = 0; i < 4; ++i) {
      int c = lane * 4 + i;
      dst[c] = src[c];
    }
  }
}

// ---------------- kernel 2: QKV GEMM per window (WMMA, TDM-staged A) ----------------
__global__ void k_qkv(const float* __restrict__ XT, const float* __restrict__ qkv_w,
                      const float* __restrict__ qkv_b, float* __restrict__ QKV) {
  __shared__ float sA[KTOK * CCH];  // 24 KB
  int m = blockIdx.x;
  stage_tile(sA, XT + (long long)m * KTOK * CCH, KTOK, CCH);
  __builtin_prefetch(qkv_w, 0, 0);
  __syncthreads();
  int wave = threadIdx.x >> 5, lane = threadIdx.x & 31;
  int l = lane & 15;
  for (int t = wave; t < 72; t += 8) {  // 3 row-tiles x 24 col-tiles
    int rt = t / 24, ct = t % 24;
    int row0 = rt * 16, n0 = ct * 16;
    float bias = qkv_b[n0 + l];
    v8f c;
#pragma unroll
    for (int r = 0; r < 8; ++r) c[r] = bias;
    const float* A = sA + row0 * CCH;
    const float* B = qkv_w + n0;
    for (int k0 = 0; k0 < CCH; k0 += 4) {
      v2f a = frag_rm(A, CCH, k0, lane);
      v2f b = frag_b(B, 3 * CCH, k0, lane);
      c = wmma4(a, b, c);
    }
    store_c(QKV + ((long long)m * KTOK + row0) * (3 * CCH) + n0, 3 * CCH, c, lane);
  }
}

// -------- kernel 3: per-window attention: S=QK^T, mask, softmax, O=PV --------
__global__ void k_attn(const float* __restrict__ QKV, float* __restrict__ O) {
  __shared__ float sS[HEADS * KTOK * KTOK];  // 36 KB scores
  int m = blockIdx.x;
  const float* qkv = QKV + (long long)m * KTOK * (3 * CCH);
  int wave = threadIdx.x >> 5, lane = threadIdx.x & 31;
  const float scale = 0.17677669529663688f;  // 32^-0.5
  // S_h = Q_h K_h^T * scale : 4 heads x 3x3 tiles
  for (int t = wave; t < HEADS * 9; t += 8) {
    int h = t / 9, rem = t % 9;
    int row0 = (rem / 3) * 16, n0 = (rem % 3) * 16;
    v8f c;
#pragma unroll
    for (int r = 0; r < 8; ++r) c[r] = 0.f;
    const float* Aq = qkv + row0 * (3 * CCH) + 96 * h;        // Q rows
    const float* Bk = qkv + n0 * (3 * CCH) + 96 * h + DHEAD;  // K rows (B^T)
    for (int k0 = 0; k0 < DHEAD; k0 += 4) {
      v2f a = frag_rm(Aq, 3 * CCH, k0, lane);
      v2f b = frag_rm(Bk, 3 * CCH, k0, lane);  // B[k][n] = K[n][k]
      c = wmma4(a, b, c);
    }
#pragma unroll
    for (int r = 0; r < 8; ++r) c[r] *= scale;
    store_c(sS + (h * KTOK + row0) * KTOK + n0, KTOK, c, lane);
  }
  __syncthreads();
  // key-padding mask (-1e4 at key>=40) + softmax, one thread per (h, query) row
  if (threadIdx.x < HEADS * KTOK) {
    float* rowp = sS + threadIdx.x * KTOK;
    for (int j = KNP; j < KTOK; ++j) rowp[j] = -10000.0f;
    float mx = rowp[0];
    for (int j = 1; j < KTOK; ++j) mx = fmaxf(mx, rowp[j]);
    float s = 0.f;
    for (int j = 0; j < KTOK; ++j) {
      float e = __expf(rowp[j] - mx);
      rowp[j] = e;
      s += e;
    }
    float inv = 1.0f / s;
    for (int j = 0; j < KTOK; ++j) rowp[j] *= inv;
  }
  __syncthreads();
  // O_h = P_h V_h : 4 heads x 3 row-tiles x 2 col-tiles
  for (int t = wave; t < HEADS * 6; t += 8) {
    int h = t / 6, rem = t % 6;
    int row0 = (rem / 2) * 16, n0 = (rem % 2) * 16;
    v8f c;
#pragma unroll
    for (int r = 0; r < 8; ++r) c[r] = 0.f;
    const float* Ap = sS + (h * KTOK + row0) * KTOK;
    const float* Bv = qkv + 96 * h + 64 + n0;  // V rows, stride 384
    for (int k0 = 0; k0 < KTOK; k0 += 4) {
      v2f a = frag_rm(Ap, KTOK, k0, lane);
      v2f b = frag_b(Bv, 3 * CCH, k0, lane);
      c = wmma4(a, b, c);
    }
    store_c(O + ((long long)m * KTOK + row0) * CCH + h * DHEAD + n0, CCH, c, lane);
  }
}

// ---- kernel 4: proj GEMM + ls1 residual + LN2 -> XA (shortcut2) & XALN ----
__global__ void k_proj(const float* __restrict__ O, const float* __restrict__ proj_w,
                       const float* __restrict__ proj_b, const float* __restrict__ XT,
                       const float* __restrict__ ls1, const float* __restrict__ g2,
                       const float* __restrict__ b2, float* __restrict__ XA,
                       float* __restrict__ XALN) {
  __shared__ float sO[KTOK * CCH];  // 24 KB
  __shared__ float sP[KTOK * CCH];  // 24 KB
  int m = blockIdx.x;
  stage_tile(sO, O + (long long)m * KTOK * CCH, KTOK, CCH);
  __syncthreads();
  int wave = threadIdx.x >> 5, lane = threadIdx.x & 31;
  int l = lane & 15;
  for (int t = wave; t < 24; t += 8) {  // 3 x 8 tiles
    int rt = t / 8, ct = t % 8;
    int row0 = rt * 16, n0 = ct * 16;
    float bias = proj_b[n0 + l];
    v8f c;
#pragma unroll
    for (int r = 0; r < 8; ++r) c[r] = bias;
    const float* A = sO + row0 * CCH;
    const float* B = proj_w + n0;
    for (int k0 = 0; k0 < CCH; k0 += 4) {
      v2f a = frag_rm(A, CCH, k0, lane);
      v2f b = frag_b(B, CCH, k0, lane);
      c = wmma4(a, b, c);
    }
    store_c(sP + row0 * CCH + n0, CCH, c, lane);
  }
  __syncthreads();
  // rows kk<40 only: xa = shortcut(XT row) + proj*ls1; store XA and LN2 -> XALN
  for (int rr = 0; rr < 5; ++rr) {
    int kk = wave * 5 + rr;  // 8 waves * 5 = 40 rows
    const float* xtr = XT + ((long long)m * KTOK + kk) * CCH;
    float v[4];
    float s = 0.f, ss = 0.f;
#pragma unroll
    for (int i = 0; i < 4; ++i) {
      int c = lane * 4 + i;
      float xa = xtr[c] + sP[kk * CCH + c] * ls1[c];
      v[i] = xa;
      s += xa;
      ss += xa * xa;
    }
    s = wsum(s);
    ss = wsum(ss);
    float mean = s * (1.0f / CCH);
    float var = ss * (1.0f / CCH) - mean * mean;
    float rstd = rsqrtf(var + 1e-5f);
    float* xar = XA + ((long long)m * KNP + kk) * CCH;
    float* xlr = XALN + ((long long)m * KNP + kk) * CCH;
#pragma unroll
    for (int i = 0; i < 4; ++i) {
      int c = lane * 4 + i;
      xar[c] = v[i];
      xlr[c] = (v[i] - mean) * rstd * g2[c] + b2[c];
    }
  }
}

// ------------- kernel 5: fused gated-GLU MLP (two WMMA GEMMs) -------------
__global__ void k_mlp(const float* __restrict__ XALN, const float* __restrict__ glu_w,
                      const float* __restrict__ glu_b, const float* __restrict__ out_w,
                      const float* __restrict__ out_b, float* __restrict__ Y) {
  __shared__ float sX[32 * CCH];     // 16 KB
  __shared__ float sG[32 * GINNER];  // 40 KB gated activations
  long long r0 = (long long)blockIdx.x * 32;
  stage_tile(sX, XALN + r0 * CCH, 32, CCH);
  __builtin_prefetch(glu_w, 0, 0);
  __syncthreads();
  int wave = threadIdx.x >> 5, lane = threadIdx.x & 31;
  int l = lane & 15;
  // phase 1: u and g tiles, gated = u * gelu(g)
  for (int t = wave; t < 40; t += 8) {  // 2 row-tiles x 20 col-tiles
    int rt = t / 20, ct = t % 20;
    int row0 = rt * 16, n0 = ct * 16;
    float bu0 = glu_b[n0 + l], bg0 = glu_b[GINNER + n0 + l];
    v8f cu, cg;
#pragma unroll
    for (int r = 0; r < 8; ++r) {
      cu[r] = bu0;
      cg[r] = bg0;
    }
    const float* A = sX + row0 * CCH;
    const float* Bu = glu_w + n0;
    const float* Bg = glu_w + GINNER + n0;
    for (int k0 = 0; k0 < CCH; k0 += 4) {
      v2f a = frag_rm(A, CCH, k0, lane);
      cu = wmma4(a, frag_b(Bu, 2 * GINNER, k0, lane), cu);
      cg = wmma4(a, frag_b(Bg, 2 * GINNER, k0, lane), cg);
    }
    v8f gt;
#pragma unroll
    for (int r = 0; r < 8; ++r) gt[r] = cu[r] * gelu_exact(cg[r]);
    store_c(sG + row0 * GINNER + n0, GINNER, gt, lane);
  }
  __syncthreads();
  // phase 2: y = gated(32x320) @ out_w(320x128) + out_b
  for (int t = wave; t < 16; t += 8) {  // 2 x 8 tiles
    int rt = t / 8, ct = t % 8;
    int row0 = rt * 16, n0 = ct * 16;
    float bias = out_b[n0 + l];
    v8f c;
#pragma unroll
    for (int r = 0; r < 8; ++r) c[r] = bias;
    const float* A = sG + row0 * GINNER;
    const float* B = out_w + n0;
    for (int k0 = 0; k0 < GINNER; k0 += 4) {
      v2f a = frag_rm(A, GINNER, k0, lane);
      c = wmma4(a, frag_b(B, CCH, k0, lane), c);
    }
    store_c(Y + (r0 + row0) * CCH + n0, CCH, c, lane);
  }
}

// -------- kernel 6: CB per-batch-group sums (group = window >> 8) --------
__global__ void k_cbsum(const float* __restrict__ Y, const int* __restrict__ iw,
                        float* __restrict__ CB) {
  int b = blockIdx.x;   // 0..7
  int c = threadIdx.x;  // 0..127
  float acc = 0.f;
  for (int m = 0; m < MSEL; ++m) {
    if ((iw[m] >> 8) == b) {
      const float* yr = Y + (long long)m * KNP * CCH + c;
      for (int kk = 0; kk < KNP; ++kk) acc += yr[kk * CCH];
    }
  }
  CB[b * CCH + c] = acc;
}

// -------- kernel 7: CB mix + ls2 residual + scatter into d_out --------
__global__ void k_final(const float* __restrict__ XA, const float* __restrict__ Y,
                        const float* __restrict__ CB, const float* __restrict__ ls2,
                        const int* __restrict__ iw, const int* __restrict__ enable_cb,
                        float* __restrict__ out) {
  long long idx = (long long)blockIdx.x * blockDim.x + threadIdx.x;
  int c = (int)(idx & (CCH - 1));
  long long row = idx >> 7;
  int m = (int)(row / KNP);
  int kk = (int)(row - (long long)m * KNP);
  int w = iw[m];
  float y = Y[idx];
  if (enable_cb[0]) {
    int b = w >> 8;                                      // w / (NWIN/BSZ)
    y = 0.5f * y + (0.5f / 16384.0f) * CB[b * CCH + c];  // mean over 256*64 slots
  }
  float o = XA[idx] + y * ls2[c];
  out[(((long long)w * PWIN) + kk) * CCH + c] = o;
}

// ------------------------------- launcher -------------------------------
extern "C" void kernel_launch(void* const* d_in, const int* in_sizes, int n_in,
                              void* d_out, int out_size, void* d_ws, size_t ws_size,
                              hipStream_t stream) {
  (void)in_sizes;
  (void)n_in;
  (void)out_size;
  (void)ws_size;
  const float* x = (const float*)d_in[0];
  const float* n1_g = (const float*)d_in[1];
  const float* n1_b = (const float*)d_in[2];
  const float* n2_g = (const float*)d_in[3];
  const float* n2_b = (const float*)d_in[4];
  const float* qkv_w = (const float*)d_in[5];
  const float* qkv_b = (const float*)d_in[6];
  const float* proj_w = (const float*)d_in[7];
  const float* proj_b = (const float*)d_in[8];
  const float* ls1_g = (const float*)d_in[9];
  const float* ls2_g = (const float*)d_in[10];
  const float* glu_w = (const float*)d_in[11];
  const float* glu_b = (const float*)d_in[12];
  const float* out_w = (const float*)d_in[13];
  const float* out_b = (const float*)d_in[14];
  const int* index_window = (const int*)d_in[15];
  const int* enable_cb = (const int*)d_in[21];
  float* out = (float*)d_out;
  float* ws = (float*)d_ws;

  // workspace layout (floats)
  const long long SZ_XT = (long long)MSEL * KTOK * CCH;       //  9.44 M
  const long long SZ_QKV = (long long)MSEL * KTOK * 3 * CCH;  // 28.31 M
  const long long SZ_O = SZ_XT;                               //  9.44 M
  const long long SZ_ROWS = (long long)MSEL * KNP * CCH;      //  7.86 M
  float* XT = ws;
  float* QKV = XT + SZ_XT;
  float* O = QKV + SZ_QKV;
  float* XA = O + SZ_O;
  float* XALN = XA + SZ_ROWS;
  float* Y = XALN + SZ_ROWS;
  float* CB = Y + SZ_ROWS;  // 8*128 floats

  dim3 blk(256);
  k_ln1<<<dim3(NWIN * PWIN / 8), blk, 0, stream>>>(x, n1_g, n1_b, out);
  k_build_xt<<<dim3(MSEL * KTOK / 8), blk, 0, stream>>>(out, index_window, n2_g, n2_b,
                                                        XT);
  k_qkv<<<dim3(MSEL), blk, 0, stream>>>(XT, qkv_w, qkv_b, QKV);
  k_attn<<<dim3(MSEL), blk, 0, stream>>>(QKV, O);
  k_proj<<<dim3(MSEL), blk, 0, stream>>>(O, proj_w, proj_b, XT, ls1_g, n2_g, n2_b, XA,
                                         XALN);
  k_mlp<<<dim3(MSEL * KNP / 32), blk, 0, stream>>>(XALN, glu_w, glu_b, out_w, out_b, Y);
  k_cbsum<<<dim3(BSZ), dim3(CCH), 0, stream>>>(Y, index_window, CB);
  k_final<<<dim3(MSEL * KNP * CCH / 256), blk, 0, stream>>>(XA, Y, CB, ls2_g,
                                                            index_window, enable_cb,
                                                            out);
}